// DSNTJSDDistanceDoubleLoss_81252191305987
// MI455X (gfx1250) — compile-verified
//
#include <hip/hip_runtime.h>
#include <cstdint>

#define W_    512
#define HW_   (512*512)
#define NCH_  128            // B*C = 64*2
#define BPC_  16             // blocks per channel
#define PERB_ (HW_/BPC_)     // 16384 elements per block
#define TILE_ 2048           // TDM tile (floats) = 8KB
#define NT_   (PERB_/TILE_)  // 8 tiles per block
#define NPART_ (NCH_*BPC_)   // 2048 partials

typedef unsigned int u32x4_t __attribute__((ext_vector_type(4)));
typedef int i32x8_t __attribute__((ext_vector_type(8)));
typedef int i32x4_t __attribute__((ext_vector_type(4)));

// ---- Tensor Data Mover: 1-D tile Global -> LDS (CDNA5 ISA section 8 D# layout) ----
__device__ __forceinline__ void tdm_load_1d(const float* gsrc, const float* ldsDst,
                                            unsigned nfloats) {
  unsigned long long ga = (unsigned long long)(uintptr_t)gsrc;
  unsigned lds = (unsigned)(uintptr_t)ldsDst;   // LDS aperture: addr[31:0] == LDS byte offset
  u32x4_t g0;
  g0.x = 1u;                                        // count=1, user descriptor
  g0.y = lds;                                       // lds_addr [63:32]
  g0.z = (unsigned)(ga & 0xFFFFFFFFull);            // global_addr lo
  g0.w = ((unsigned)((ga >> 32) & 0x01FFFFFFull))   // global_addr [56:32]
       | (2u << 30);                                // type=2 ("image")
  i32x8_t g1;
  g1[0] = (int)(2u << 16);                          // data_size=4B, no multicast
  g1[1] = (int)((nfloats & 0xFFFFu) << 16);         // tensor_dim0 lo16 (bits 79:48)
  g1[2] = (int)(((nfloats >> 16) & 0xFFFFu) | (1u << 16)); // tensor_dim0 hi16 | tensor_dim1=1 lo16
  g1[3] = (int)((nfloats & 0xFFFFu) << 16);         // tensor_dim1 hi16=0 | tile_dim0
  g1[4] = 1;                                        // tile_dim1=1, tile_dim2=0
  g1[5] = (int)nfloats;                             // tensor_dim0_stride lo32
  g1[6] = 0;                                        // stride hi16 | tensor_dim1_stride lo16
  g1[7] = 0;
  i32x4_t z4; z4[0]=0; z4[1]=0; z4[2]=0; z4[3]=0;   // groups 2/3 unused (<=2D)
  i32x8_t z8; z8[0]=0; z8[1]=0; z8[2]=0; z8[3]=0; z8[4]=0; z8[5]=0; z8[6]=0; z8[7]=0;
  // clang-23 / therock-10.0 form: 6 args (g0, g1, g2, g3, extra, cpol)
  __builtin_amdgcn_tensor_load_to_lds(g0, g1, z4, z4, z8, 0);
}

// ---- Pass 1: per-slice softmax stats of input + argmax of target ----
__global__ __launch_bounds__(256) void kA(const float* __restrict__ inp,
                                          const float* __restrict__ tgt,
                                          float* __restrict__ aMax, float* __restrict__ aSum,
                                          float* __restrict__ aTv, int* __restrict__ aTi) {
  const int tid  = threadIdx.x;
  const int ch   = blockIdx.x >> 4;
  const int blk  = blockIdx.x & (BPC_ - 1);
  const long long cbase = (long long)ch * HW_;
  const int bbase = blk * PERB_;
  const float4* in4 = (const float4*)(inp + cbase + bbase);
  const float4* tg4 = (const float4*)(tgt + cbase + bbase);

  float m = -3.4e38f;
  #pragma unroll
  for (int j = 0; j < 16; ++j) {
    float4 v = in4[tid + j*256];
    m = fmaxf(m, fmaxf(fmaxf(v.x, v.y), fmaxf(v.z, v.w)));
  }
  float s = 0.f;
  #pragma unroll
  for (int j = 0; j < 16; ++j) {        // second read hits WGP$/L2
    float4 v = in4[tid + j*256];
    s += __expf(v.x - m) + __expf(v.y - m) + __expf(v.z - m) + __expf(v.w - m);
  }
  float tv = -3.4e38f; int ti = 0;
  #pragma unroll
  for (int j = 0; j < 16; ++j) {
    float4 v = tg4[tid + j*256];
    int e = bbase + (tid + j*256)*4;
    if (v.x > tv) { tv = v.x; ti = e;   }
    if (v.y > tv) { tv = v.y; ti = e+1; }
    if (v.z > tv) { tv = v.z; ti = e+2; }
    if (v.w > tv) { tv = v.w; ti = e+3; }
  }
  __shared__ float rm[256], rs[256], rtv[256];
  __shared__ int rti[256];
  rm[tid]=m; rs[tid]=s; rtv[tid]=tv; rti[tid]=ti;
  __syncthreads();
  for (int st = 128; st > 0; st >>= 1) {
    if (tid < st) {
      float m2 = rm[tid+st], s2 = rs[tid+st];
      float M = fmaxf(rm[tid], m2);
      rs[tid] = rs[tid]*__expf(rm[tid]-M) + s2*__expf(m2-M);
      rm[tid] = M;
      float v2 = rtv[tid+st]; int i2 = rti[tid+st];
      if (v2 > rtv[tid] || (v2 == rtv[tid] && i2 < rti[tid])) { rtv[tid]=v2; rti[tid]=i2; }
    }
    __syncthreads();
  }
  if (tid == 0) {
    aMax[blockIdx.x] = rm[0];
    aSum[blockIdx.x] = rs[0];
    aTv [blockIdx.x] = rtv[0];
    aTi [blockIdx.x] = rti[0];
  }
}

// ---- Pass 2: merge per-channel stats ----
__global__ __launch_bounds__(128) void kB(const float* __restrict__ aMax,
                                          const float* __restrict__ aSum,
                                          const float* __restrict__ aTv,
                                          const int*   __restrict__ aTi,
                                          float* M_, float* invS_, float* tx_, float* ty_) {
  int c = threadIdx.x;
  float M = -3.4e38f;
  for (int k = 0; k < BPC_; ++k) M = fmaxf(M, aMax[c*BPC_ + k]);
  float S = 0.f;
  for (int k = 0; k < BPC_; ++k) S += aSum[c*BPC_ + k] * __expf(aMax[c*BPC_ + k] - M);
  float tv = -3.4e38f; int ti = 0;
  for (int k = 0; k < BPC_; ++k) {
    float v = aTv[c*BPC_ + k]; int i = aTi[c*BPC_ + k];
    if (v > tv || (v == tv && i < ti)) { tv = v; ti = i; }
  }
  M_[c]    = M;
  invS_[c] = 1.f / S;
  tx_[c]   = (float)((ti & (W_ - 1)) + 1) * (1.f / (float)W_);
  ty_[c]   = (float)((ti >> 9) + 1) * (1.f / 512.f);
}

// ---- Pass 3: TDM-staged streaming pass: px, py, JSD partials ----
__global__ __launch_bounds__(256) void kC(const float* __restrict__ inp,
                                          const float* __restrict__ tgt,
                                          const float* __restrict__ M_,
                                          const float* __restrict__ invS_,
                                          float* __restrict__ pPx, float* __restrict__ pPy,
                                          float* __restrict__ pJs) {
  __shared__ __align__(16) float bufI[2][TILE_];
  __shared__ __align__(16) float bufT[2][TILE_];
  __shared__ float r0[256], r1[256], r2[256];
  const int tid  = threadIdx.x;
  const int ch   = blockIdx.x >> 4;
  const int blk  = blockIdx.x & (BPC_ - 1);
  const long long cbase = (long long)ch * HW_;
  const int bbase = blk * PERB_;
  const float* gin = inp + cbase + bbase;
  const float* gtg = tgt + cbase + bbase;
  const float M = M_[ch], invS = invS_[ch];

  if (tid < 32) {                                   // wave 0 drives the TDM pipeline
    tdm_load_1d(gin, &bufI[0][0], TILE_);
    tdm_load_1d(gtg, &bufT[0][0], TILE_);
  }
  float px = 0.f, py = 0.f, js = 0.f;
  for (int it = 0; it < NT_; ++it) {
    const int cur = it & 1;
    if (tid < 32) {
      if (it + 1 < NT_) {                           // prefetch next tile pair, keep it in flight
        tdm_load_1d(gin + (it+1)*TILE_, &bufI[cur^1][0], TILE_);
        tdm_load_1d(gtg + (it+1)*TILE_, &bufT[cur^1][0], TILE_);
        __builtin_amdgcn_s_wait_tensorcnt(2);       // current pair done; next pair outstanding
      } else {
        __builtin_amdgcn_s_wait_tensorcnt(0);
      }
    }
    __syncthreads();
    #pragma unroll
    for (int j = 0; j < TILE_/1024; ++j) {          // 2 x float4 per thread per tile
      int q = tid + j*256;
      float4 vi = ((const float4*)&bufI[cur][0])[q];
      float4 vt = ((const float4*)&bufT[cur][0])[q];
      int g = bbase + it*TILE_ + q*4;               // flat index within channel
      float p0 = __expf(vi.x - M) * invS;
      float p1 = __expf(vi.y - M) * invS;
      float p2 = __expf(vi.z - M) * invS;
      float p3 = __expf(vi.w - M) * invS;
      float w0 = (float)((g & (W_ - 1)) + 1);
      float yh = (float)((g >> 9) + 1);
      px += p0*w0 + p1*(w0+1.f) + p2*(w0+2.f) + p3*(w0+3.f);
      py += (p0 + p1 + p2 + p3) * yh;
      float m0 = 0.5f*(vt.x + p0), m1 = 0.5f*(vt.y + p1);
      float m2 = 0.5f*(vt.z + p2), m3 = 0.5f*(vt.w + p3);
      js += m0*__logf(m0) - m0*m0;
      js += m1*__logf(m1) - m1*m1;
      js += m2*__logf(m2) - m2*m2;
      js += m3*__logf(m3) - m3*m3;
    }
    __syncthreads();                                // everyone done before buffer is re-DMA'd
  }
  r0[tid]=px; r1[tid]=py; r2[tid]=js;
  __syncthreads();
  for (int st = 128; st > 0; st >>= 1) {
    if (tid < st) { r0[tid]+=r0[tid+st]; r1[tid]+=r1[tid+st]; r2[tid]+=r2[tid+st]; }
    __syncthreads();
  }
  if (tid == 0) { pPx[blockIdx.x]=r0[0]; pPy[blockIdx.x]=r1[0]; pJs[blockIdx.x]=r2[0]; }
}

// ---- Pass 4: epilogue ----
__global__ __launch_bounds__(128) void kD(const float* __restrict__ pPx,
                                          const float* __restrict__ pPy,
                                          const float* __restrict__ pJs,
                                          const float* __restrict__ tx_,
                                          const float* __restrict__ ty_,
                                          float* __restrict__ out) {
  __shared__ float lpx[128], lpy[128], lval[128], lpair[64];
  int c = threadIdx.x;
  float sx=0.f, sy=0.f, sj=0.f;
  for (int k = 0; k < BPC_; ++k) { sx+=pPx[c*BPC_+k]; sy+=pPy[c*BPC_+k]; sj+=pJs[c*BPC_+k]; }
  float px = sx * (1.f/512.f);
  float py = sy * (1.f/512.f);
  float jsd = sj * (1.f/(float)HW_);
  float dx = tx_[c]-px, dy = ty_[c]-py;
  lpx[c]=px; lpy[c]=py; lval[c]=sqrtf(dx*dx+dy*dy)+jsd;
  __syncthreads();
  if (c < 64) {
    int c0 = 2*c, c1 = 2*c+1;
    float pdx = lpx[c0]-lpx[c1], pdy = lpy[c0]-lpy[c1];
    float pd  = sqrtf(pdx*pdx + pdy*pdy);
    float tdx = tx_[c0]-tx_[c1], tdy = ty_[c0]-ty_[c1];
    float td  = sqrtf(tdx*tdx + tdy*tdy);
    lpair[c]  = fabsf(pd - td);
  }
  __syncthreads();
  for (int st = 64; st > 0; st >>= 1) { if (c < st) lval[c]  += lval[c+st];  __syncthreads(); }
  for (int st = 32; st > 0; st >>= 1) { if (c < st) lpair[c] += lpair[c+st]; __syncthreads(); }
  if (c == 0) out[0] = (lval[0] + lpair[0]) * (1.f/64.f);
}

extern "C" void kernel_launch(void* const* d_in, const int* in_sizes, int n_in,
                              void* d_out, int out_size, void* d_ws, size_t ws_size,
                              hipStream_t stream) {
  const float* inp = (const float*)d_in[0];
  const float* tgt = (const float*)d_in[1];
  float* out = (float*)d_out;
  float* ws  = (float*)d_ws;
  float* aMax = ws;                 // [2048]
  float* aSum = ws + 2048;          // [2048]
  float* aTv  = ws + 4096;          // [2048]
  int*   aTi  = (int*)(ws + 6144);  // [2048]
  float* pPx  = ws + 8192;          // [2048]
  float* pPy  = ws + 10240;         // [2048]
  float* pJs  = ws + 12288;         // [2048]
  float* M_   = ws + 14336;         // [128]
  float* invS = ws + 14464;         // [128]
  float* tx   = ws + 14592;         // [128]
  float* ty   = ws + 14720;         // [128]

  kA<<<NPART_, 256, 0, stream>>>(inp, tgt, aMax, aSum, aTv, aTi);
  kB<<<1, 128, 0, stream>>>(aMax, aSum, aTv, aTi, M_, invS, tx, ty);
  kC<<<NPART_, 256, 0, stream>>>(inp, tgt, M_, invS, pPx, pPy, pJs);
  kD<<<1, 128, 0, stream>>>(pPx, pPy, pJs, tx, ty, out);
}